// MyModel_87522843560682
// MI455X (gfx1250) — compile-verified
//
#include <hip/hip_runtime.h>

#define MDIM 4096
#define NB   64          // stage block size (64 stages)

typedef __attribute__((ext_vector_type(2))) float v2f;
typedef __attribute__((ext_vector_type(8))) float v8f;

// ---------------------------------------------------------------------------
// Zero the accumulator panel acc[M][16] in workspace.
// ---------------------------------------------------------------------------
__global__ void zero_acc_kernel(float* __restrict__ acc, int n) {
    int i = blockIdx.x * blockDim.x + threadIdx.x;
    if (i < n) acc[i] = 0.0f;
}

// ---------------------------------------------------------------------------
// Stage k diagonal-block solve: 64 threads, sequential forward substitution
// over the NB x NB diagonal block. rhs comes from b minus the accumulated
// contribution of all previously-solved column panels (acc column 0; all 16
// acc columns are identical by construction).
// ---------------------------------------------------------------------------
__global__ void solve_block_kernel(const float* __restrict__ L,
                                   const float* __restrict__ b,
                                   const float* __restrict__ acc,
                                   float* __restrict__ x, int k) {
    const int i0  = k * NB;
    const int tid = threadIdx.x;                    // 0..NB-1
    __shared__ float rhs[NB];
    __shared__ float xs[NB];
    rhs[tid] = b[i0 + tid] - acc[(size_t)(i0 + tid) * 16];
    __syncthreads();
    for (int i = 0; i < NB; ++i) {
        if (tid == i)
            xs[i] = rhs[i] / L[(size_t)(i0 + i) * MDIM + (i0 + i)];
        __syncthreads();
        if (tid > i)
            rhs[tid] -= L[(size_t)(i0 + tid) * MDIM + (i0 + i)] * xs[i];
        __syncthreads();
    }
    x[i0 + tid] = xs[tid];
}

// ---------------------------------------------------------------------------
// Stage k right-looking panel update via V_WMMA_F32_16X16X4_F32.
// Each wave owns a 16-row tile of the remaining rows; B carries the just-
// solved x-block replicated across all 16 columns, so
//   acc[r][0..15] += sum_{j in block k} L[r][j] * x[j]
// is computed as a chain of 16 fp32 WMMAs (K = 64 in steps of 4).
// blockDim = 128 (4 waves -> 64 rows per workgroup).
//
// A layout (32-bit A 16x4, ISA 7.12.2): lanes 0-15 hold K=0 (vgpr0), K=1
// (vgpr1); lanes 16-31 hold K=2, K=3.  D layout: vgpr v -> M = v + (lane<16 ?
// 0 : 8), N = lane & 15.
// ---------------------------------------------------------------------------
__global__ void update_panel_kernel(const float* __restrict__ L,
                                    const float* __restrict__ x,
                                    float* __restrict__ acc, int k) {
    const int j0      = k * NB;
    const int lane    = threadIdx.x & 31;
    const int wave    = threadIdx.x >> 5;
    const int rowBase = j0 + NB + blockIdx.x * 64 + wave * 16;
    const int mrow    = lane & 15;
    const int khalf   = (lane >> 4) << 1;           // 0 or 2

    const float* __restrict__ Lrow = L + (size_t)(rowBase + mrow) * MDIM + j0;
    const float* __restrict__ xblk = x + j0;

    v8f c = {};
#pragma unroll
    for (int kk = 0; kk < NB; kk += 4) {
        v2f a, bx;
        a.x  = Lrow[kk + khalf];
        a.y  = Lrow[kk + khalf + 1];
        bx.x = xblk[kk + khalf];                    // 16 replicated RHS columns
        bx.y = xblk[kk + khalf + 1];
        __builtin_prefetch(Lrow + kk + 16, 0, 3);   // global_prefetch_b8
        c = __builtin_amdgcn_wmma_f32_16x16x4_f32(
                /*neg_a=*/false, a, /*neg_b=*/false, bx,
                /*c_mod=*/(short)0, c, /*reuse_a=*/false, /*reuse_b=*/false);
    }

    const int col  = lane & 15;
    const int mofs = (lane < 16) ? 0 : 8;
#pragma unroll
    for (int v = 0; v < 8; ++v) {
        size_t idx = (size_t)(rowBase + mofs + v) * 16 + col;
        acc[idx] += c[v];                           // one wave per row-tile: race-free
    }
}

// ---------------------------------------------------------------------------
// Broadcast x to the four solution slots and set the four boolean flags true.
// Output layout: [sol_inline | sol_precomputed | sol_inv | sol_naive | c1 c2 c3 all]
// ---------------------------------------------------------------------------
__global__ void write_out_kernel(const float* __restrict__ x,
                                 float* __restrict__ out) {
    int i = blockIdx.x * blockDim.x + threadIdx.x;
    if (i < MDIM) {
        float v = x[i];
        out[i]            = v;
        out[MDIM + i]     = v;
        out[2 * MDIM + i] = v;
        out[3 * MDIM + i] = v;
    }
    if (i < 4) out[4 * MDIM + i] = 1.0f;
}

// ---------------------------------------------------------------------------
extern "C" void kernel_launch(void* const* d_in, const int* in_sizes, int n_in,
                              void* d_out, int out_size, void* d_ws, size_t ws_size,
                              hipStream_t stream) {
    const float* L = (const float*)d_in[0];   // 4096 x 4096 row-major fp32
    const float* b = (const float*)d_in[1];   // 4096 fp32
    float* out = (float*)d_out;

    float* x   = (float*)d_ws;                // M floats
    float* acc = x + MDIM;                    // M x 16 floats

    zero_acc_kernel<<<(MDIM * 16 + 255) / 256, 256, 0, stream>>>(acc, MDIM * 16);

    const int nblk = MDIM / NB;               // 64 serialized stages
    for (int k = 0; k < nblk; ++k) {
        solve_block_kernel<<<1, NB, 0, stream>>>(L, b, acc, x, k);
        int rows = MDIM - (k + 1) * NB;       // rows still to update
        if (rows > 0)
            update_panel_kernel<<<rows / 64, 128, 0, stream>>>(L, x, acc, k);
    }

    write_out_kernel<<<(MDIM + 255) / 256, 256, 0, stream>>>(x, out);
}